// Qwen3NextGatedDeltaNet_2482491097101
// MI455X (gfx1250) — compile-verified
//
#include <hip/hip_runtime.h>
#include <hip/hip_bf16.h>

typedef __attribute__((ext_vector_type(2))) float v2f;
typedef __attribute__((ext_vector_type(8))) float v8f;

#define B_   8
#define T_   128
#define H_   16
#define HV_  32
#define K_   128
#define V_   128
#define CHK  16
#define NCHUNK (T_/CHK)
#define STR  132          // LDS row stride (floats): 132 mod 64 = 4 -> conflict-free half-wave scatter
#define EPSF 1e-6f
#define RSQRT_K 0.08838834764831845f   // 128^-0.5

__device__ __forceinline__ v8f wmma4(v2f a, v2f b, v8f c) {
  // D = A(16x4,f32) * B(4x16,f32) + C(16x16,f32)
  return __builtin_amdgcn_wmma_f32_16x16x4_f32(false, a, false, b, (short)0, c, false, false);
}

__global__ __launch_bounds__(256, 1)
void gdn_fwd_kernel(const float* __restrict__ qg, const float* __restrict__ kg,
                    const float* __restrict__ vg, const float* __restrict__ ag,
                    const float* __restrict__ bg, const float* __restrict__ Alog,
                    const float* __restrict__ dtb, const float* __restrict__ h0,
                    const float* __restrict__ zg, const float* __restrict__ nw,
                    float* __restrict__ yout, float* __restrict__ sout)
{
  __shared__ float Sm[K_][STR];      // state mirror (B-operand source)
  __shared__ float kn_s[CHK][STR];   // l2-normalized k
  __shared__ float kt_s[CHK][STR];   // k~ = kn / Lam_t
  __shared__ float qh_s[CHK][STR];   // q^ = qn * scale * Lam_t
  __shared__ float vb_s[CHK][STR];   // beta*v  -> becomes RHS0 in place
  __shared__ float U0_s[CHK][STR];   // K @ S0
  __shared__ float R_s [CHK][STR];   // WY "u" rows
  __shared__ float Bm_s[CHK][17];    // strict lower (I+B) matrix
  __shared__ float Tm_s[CHK][17];    // (I+B)^-1
  __shared__ float Aq_s[CHK][17];    // masked q^.k~^T
  __shared__ float red1[CHK][16];
  __shared__ float red2[CHK][16];
  __shared__ float rowg[CHK], rowb[CHK], rowL[CHK], rowIL[CHK], rowBL[CHK];
  __shared__ float rowQn[CHK], rowKn[CHK];
  __shared__ float rowSS[CHK];
  __shared__ float LamC_s;

  const int tid  = threadIdx.x;
  const int wave = tid >> 5;
  const int lane = tid & 31;
  const int l16  = lane & 15;
  const int hi   = lane >> 4;

  const int bid = blockIdx.x;
  const int bb  = bid / HV_;
  const int hv  = bid % HV_;
  const int hh  = hv / (HV_ / H_);   // GQA repeat: H -> HV

  // ---- initial state into per-wave accumulator fragments (wave w owns K-rows 16w..16w+15) ----
  v8f S[8];
  {
    const float* hp = h0 + (((size_t)bb * HV_ + hv) * K_) * (size_t)V_;
    #pragma unroll
    for (int tv = 0; tv < 8; ++tv) {
      v8f c = {};
      #pragma unroll
      for (int j = 0; j < 8; ++j) {
        int r = 16*wave + j + 8*hi;
        c[j] = hp[(size_t)r * V_ + 16*tv + l16];
      }
      S[tv] = c;
    }
  }

  const int row = tid >> 4;   // 0..15 : chunk row
  const int sub = tid & 15;   // 16 threads per row, 8 elems each

  for (int ch = 0; ch < NCHUNK; ++ch) {
    const int t0 = ch * CHK;

    // ================= phase A: gates, l2norm, staging =================
    if (tid < CHK) {
      int t = t0 + tid;
      float av = ag[((size_t)bb*T_ + t)*HV_ + hv] + dtb[hv];
      float sp = (av > 20.f) ? av : log1pf(__expf(av));
      rowg[tid] = -__expf(Alog[hv]) * sp;
      float bv = bg[((size_t)bb*T_ + t)*HV_ + hv];
      rowb[tid] = 1.f / (1.f + __expf(-bv));
    }
    float qv[8], kv[8];
    {
      int t = t0 + row;
      const float* qr = qg + (((size_t)bb*T_ + t)*H_ + hh)*(size_t)K_ + sub*8;
      const float* kr = kg + (((size_t)bb*T_ + t)*H_ + hh)*(size_t)K_ + sub*8;
      float s1 = 0.f, s2 = 0.f;
      #pragma unroll
      for (int i = 0; i < 8; ++i) {
        qv[i] = qr[i]; kv[i] = kr[i];
        s1 += qv[i]*qv[i]; s2 += kv[i]*kv[i];
      }
      red1[row][sub] = s1; red2[row][sub] = s2;
    }
    __syncthreads();
    if (tid == 0) {                       // cumulative log-decay
      float cg = 0.f;
      #pragma unroll
      for (int i = 0; i < CHK; ++i) {
        cg += rowg[i];
        float gc = fmaxf(cg, -60.f);      // clamp for finite exp ratios
        rowL[i]  = __expf(gc);
        rowIL[i] = __expf(-gc);
        rowBL[i] = rowb[i] * rowL[i];
      }
      LamC_s = rowL[CHK-1];
    }
    if (tid >= 32 && tid < 48) {          // row l2-norm reductions
      int r = tid - 32;
      float s1 = 0.f, s2 = 0.f;
      #pragma unroll
      for (int i = 0; i < 16; ++i) { s1 += red1[r][i]; s2 += red2[r][i]; }
      rowQn[r] = rsqrtf(s1 + EPSF) * RSQRT_K;
      rowKn[r] = rsqrtf(s2 + EPSF);
    }
    __syncthreads();
    {
      float rq = rowQn[row], rk = rowKn[row];
      float lam = rowL[row], il = rowIL[row], bt = rowb[row];
      int t = t0 + row;
      const float* vr = vg + (((size_t)bb*T_ + t)*HV_ + hv)*(size_t)V_ + sub*8;
      #pragma unroll
      for (int i = 0; i < 8; ++i) {
        int c = sub*8 + i;
        float kk = kv[i]*rk;
        kn_s[row][c] = kk;
        kt_s[row][c] = kk * il;
        qh_s[row][c] = qv[i]*rq*lam;
        vb_s[row][c] = bt * vr[i];
      }
    }
    // mirror current state to LDS (C-frag layout scatter)
    #pragma unroll
    for (int tv = 0; tv < 8; ++tv) {
      #pragma unroll
      for (int j = 0; j < 8; ++j)
        Sm[16*wave + j + 8*hi][16*tv + l16] = S[tv][j];
    }
    __syncthreads();

    // ============ phase B1: U0 = kn@S0 ; O_inter = q^@S0 ; Aq ; Bm ============
    const int cb = 16*wave;
    v8f u = {}; v8f oi = {};
    for (int k0 = 0; k0 < K_; k0 += 4) {
      v2f bS; bS.x = Sm[k0 + 2*hi][cb + l16]; bS.y = Sm[k0 + 2*hi + 1][cb + l16];
      v2f aK; aK.x = kn_s[l16][k0 + 2*hi];    aK.y = kn_s[l16][k0 + 2*hi + 1];
      v2f aQ; aQ.x = qh_s[l16][k0 + 2*hi];    aQ.y = qh_s[l16][k0 + 2*hi + 1];
      u  = wmma4(aK, bS, u);
      oi = wmma4(aQ, bS, oi);
    }
    #pragma unroll
    for (int j = 0; j < 8; ++j)
      U0_s[j + 8*hi][cb + l16] = u[j];

    if (wave == 0) {                       // Aq = q^ @ k~^T, causal mask incl diag
      v8f aq = {};
      for (int k0 = 0; k0 < K_; k0 += 4) {
        v2f aQ; aQ.x = qh_s[l16][k0+2*hi]; aQ.y = qh_s[l16][k0+2*hi+1];
        v2f bK; bK.x = kt_s[l16][k0+2*hi]; bK.y = kt_s[l16][k0+2*hi+1];
        aq = wmma4(aQ, bK, aq);
      }
      #pragma unroll
      for (int j = 0; j < 8; ++j) {
        int m = j + 8*hi;
        Aq_s[m][l16] = (l16 <= m) ? aq[j] : 0.f;
      }
    } else if (wave == 1) {                // Bm = strict_lower( (beta*Lam*kn) @ k~^T )
      v8f ak = {};
      float bl = rowBL[l16];
      for (int k0 = 0; k0 < K_; k0 += 4) {
        v2f aB; aB.x = bl*kn_s[l16][k0+2*hi]; aB.y = bl*kn_s[l16][k0+2*hi+1];
        v2f bK; bK.x = kt_s[l16][k0+2*hi];    bK.y = kt_s[l16][k0+2*hi+1];
        ak = wmma4(aB, bK, ak);
      }
      #pragma unroll
      for (int j = 0; j < 8; ++j) {
        int m = j + 8*hi;
        Bm_s[m][l16] = (l16 < m) ? ak[j] : 0.f;
      }
    }
    __syncthreads();

    // ============ phase B2: RHS0, triangular inverse ============
    #pragma unroll
    for (int e = 0; e < 8; ++e) {          // RHS0 = beta*v - beta*Lam*U0 (in place)
      int idx = tid*8 + e;
      int r = idx >> 7, c = idx & 127;
      vb_s[r][c] -= rowBL[r] * U0_s[r][c];
    }
    if (tid < 16) {
      #pragma unroll
      for (int i = 0; i < 16; ++i) Tm_s[i][tid] = (i == tid) ? 1.f : 0.f;
    }
    __syncthreads();
    if (tid < 16) {                        // column-parallel forward substitution (lane-private column)
      for (int i = tid+1; i < 16; ++i) {
        float s = 0.f;
        for (int sdx = tid; sdx < i; ++sdx) s += Bm_s[i][sdx] * Tm_s[sdx][tid];
        Tm_s[i][tid] = -s;
      }
    }
    __syncthreads();

    // ============ phase B3: R = Tm @ RHS0 ============
    v8f r8 = {};
    for (int k0 = 0; k0 < 16; k0 += 4) {
      v2f aT; aT.x = Tm_s[l16][k0+2*hi];      aT.y = Tm_s[l16][k0+2*hi+1];
      v2f bV; bV.x = vb_s[k0+2*hi][cb + l16]; bV.y = vb_s[k0+2*hi+1][cb + l16];
      r8 = wmma4(aT, bV, r8);
    }
    #pragma unroll
    for (int j = 0; j < 8; ++j)
      R_s[j + 8*hi][cb + l16] = r8[j];
    __syncthreads();

    // ============ phase B4: O = O_inter + mask(Aq) @ R ; gated RMSNorm ; y ============
    for (int k0 = 0; k0 < 16; k0 += 4) {
      v2f aA; aA.x = Aq_s[l16][k0+2*hi];     aA.y = Aq_s[l16][k0+2*hi+1];
      v2f bR; bR.x = R_s[k0+2*hi][cb + l16]; bR.y = R_s[k0+2*hi+1][cb + l16];
      oi = wmma4(aA, bR, oi);
    }
    if (tid < 16) rowSS[tid] = 0.f;
    __syncthreads();
    #pragma unroll
    for (int j = 0; j < 8; ++j) {
      float x = oi[j];
      float ss = x*x;
      ss += __shfl_xor(ss, 1);
      ss += __shfl_xor(ss, 2);
      ss += __shfl_xor(ss, 4);
      ss += __shfl_xor(ss, 8);
      if (l16 == 0) atomicAdd(&rowSS[j + 8*hi], ss);
    }
    __syncthreads();
    #pragma unroll
    for (int j = 0; j < 8; ++j) {
      int m = j + 8*hi;
      int t = t0 + m;
      int c = cb + l16;
      float inv = rsqrtf(rowSS[m] * (1.f/(float)V_) + EPSF);
      size_t off = (((size_t)bb*T_ + t)*HV_ + hv)*(size_t)V_ + c;
      float zz  = zg[off];
      float sil = zz / (1.f + __expf(-zz));
      yout[off] = oi[j] * inv * nw[c] * sil;
    }

    // ============ phase B5: S = LamC*S + (LamC*k~)^T @ R ============
    float lamC = LamC_s;
    #pragma unroll
    for (int tv = 0; tv < 8; ++tv) {
      int cb2 = 16*tv;
      v8f c = S[tv];
      #pragma unroll
      for (int j = 0; j < 8; ++j) c[j] *= lamC;
      for (int k0 = 0; k0 < 16; k0 += 4) {
        v2f aK2; aK2.x = lamC * kt_s[k0+2*hi][16*wave + l16];
                 aK2.y = lamC * kt_s[k0+2*hi+1][16*wave + l16];
        v2f bR2; bR2.x = R_s[k0+2*hi][cb2 + l16]; bR2.y = R_s[k0+2*hi+1][cb2 + l16];
        c = wmma4(aK2, bR2, c);
      }
      S[tv] = c;
    }
    __syncthreads();
  }

  // ---- final state store ----
  {
    float* sp = sout + (((size_t)bb * HV_ + hv) * K_) * (size_t)V_;
    #pragma unroll
    for (int tv = 0; tv < 8; ++tv) {
      #pragma unroll
      for (int j = 0; j < 8; ++j)
        sp[(size_t)(16*wave + j + 8*hi) * V_ + 16*tv + l16] = S[tv][j];
    }
  }
}

extern "C" void kernel_launch(void* const* d_in, const int* in_sizes, int n_in,
                              void* d_out, int out_size, void* d_ws, size_t ws_size,
                              hipStream_t stream) {
  const float* q    = (const float*)d_in[0];
  const float* k    = (const float*)d_in[1];
  const float* v    = (const float*)d_in[2];
  const float* a    = (const float*)d_in[3];
  const float* b    = (const float*)d_in[4];
  const float* Alog = (const float*)d_in[5];
  const float* dtb  = (const float*)d_in[6];
  const float* h0   = (const float*)d_in[7];
  const float* z    = (const float*)d_in[8];
  const float* nw   = (const float*)d_in[9];

  float* yout = (float*)d_out;                                   // [B,T,HV,V]
  float* sout = yout + (size_t)B_ * T_ * HV_ * V_;               // [B,HV,K,V]

  dim3 grid(B_ * HV_);
  dim3 blk(256);
  gdn_fwd_kernel<<<grid, blk, 0, stream>>>(q, k, v, a, b, Alog, dtb, h0, z, nw, yout, sout);
}